// GNNLayer_4020089389071
// MI455X (gfx1250) — compile-verified
//
#include <hip/hip_runtime.h>

// ---------------------------------------------------------------------------
// GNN layer for MI455X (gfx1250, wave32).
// out = LReLU( A*v + Bm*row_mean + Cm*col_mean + d_b ),  all matrices folded.
// Matrix math via V_WMMA_F32_16X16X4_F32 (exact fp32 WMMA).
// ---------------------------------------------------------------------------

typedef __attribute__((ext_vector_type(2))) float v2f;
typedef __attribute__((ext_vector_type(8))) float v8f;

namespace {
constexpr int       Bn   = 4;
constexpr long long En   = 262144;
constexpr int       Dn   = 64;
constexpr int       NSEG = 10000;

// workspace layout, in floats
constexpr size_t SEG_ROW = 0;                                  // B*NSEG*64 (-> Brow table)
constexpr size_t SEG_COL = SEG_ROW + (size_t)Bn * NSEG * Dn;   // B*NSEG*64 (-> Ccol table)
constexpr size_t CNT_ROW = SEG_COL + (size_t)Bn * NSEG * Dn;   // B*NSEG
constexpr size_t CNT_COL = CNT_ROW + (size_t)Bn * NSEG;        // B*NSEG
constexpr size_t AMAT    = CNT_COL + (size_t)Bn * NSEG;        // 64*64  (A  = Wo1*Wself)
constexpr size_t BMAT    = AMAT + 64 * 64;                     // 64*64  (Bm = Wo2*Wrow)
constexpr size_t CMAT    = BMAT + 64 * 64;                     // 64*64  (Cm = Wo3*Wcol)
constexpr size_t DVEC    = CMAT + 64 * 64;                     // B*64   (per-batch bias)
constexpr size_t WS_FLOATS = DVEC + (size_t)Bn * 64;
} // namespace

// guaranteed-native fp32 atomic add (L2/agent scope, relaxed)
__device__ __forceinline__ void atomAddF(float* p, float v) {
  __hip_atomic_fetch_add(p, v, __ATOMIC_RELAXED, __HIP_MEMORY_SCOPE_AGENT);
}

// ---------------------------------------------------------------------------
// Wave-level 16x64 tile transform: acc(16x64) += X(16x64) * M^T(64x64)
// X, M are LDS row-major with row stride 68 floats (bank-conflict-free).
// A-frag: lane l holds X[l&15][4*kb + 2*(l>>4) + j], j=0,1        (16x4 f32)
// B-frag: lane l holds M[16*nt + (l&15)][4*kb + 2*(l>>4) + j]     (4x16 f32)
// C/D:    lane l, vgpr j = out[m][n], m = j + 8*(l>>4), n = 16*nt + (l&15)
// ---------------------------------------------------------------------------
__device__ __forceinline__ void tile_matmul_16x64(const float (*X)[68],
                                                  const float (*M)[68],
                                                  int lane, v8f acc[4]) {
  const int r    = lane & 15;
  const int koff = (lane >> 4) << 1;
#pragma unroll
  for (int kb = 0; kb < 16; ++kb) {
    const int k0 = kb * 4 + koff;
    v2f a;
    a.x = X[r][k0];
    a.y = X[r][k0 + 1];
#pragma unroll
    for (int nt = 0; nt < 4; ++nt) {
      v2f bb;
      bb.x = M[nt * 16 + r][k0];
      bb.y = M[nt * 16 + r][k0 + 1];
      acc[nt] = __builtin_amdgcn_wmma_f32_16x16x4_f32(
          false, a, false, bb, (short)0, acc[nt], false, false);
    }
  }
}

// ---------------------------------------------------------------------------
__global__ void k_zero(float* ws, size_t n) {
  size_t i = (size_t)blockIdx.x * blockDim.x + threadIdx.x;
  size_t step = (size_t)gridDim.x * blockDim.x;
  for (; i < n; i += step) ws[i] = 0.0f;
}

// A = Wo1*Wself, Bm = Wo2*Wrow, Cm = Wo3*Wcol  (3*64*64 entries, dot-64 each)
__global__ void k_fold_mats(const float* __restrict__ W_out,
                            const float* __restrict__ W_self,
                            const float* __restrict__ W_row,
                            const float* __restrict__ W_col,
                            float* __restrict__ ws) {
  int id = blockIdx.x * blockDim.x + threadIdx.x;
  if (id >= 3 * 4096) return;
  const int which = id / 4096;
  const int rem   = id % 4096;
  const int o = rem / 64, d = rem % 64;
  const float* Wk = (which == 0) ? W_self : (which == 1) ? W_row : W_col;
  const int koff  = which * 64; // column block of W_out
  float s = 0.0f;
#pragma unroll 8
  for (int k = 0; k < 64; ++k)
    s += W_out[o * 256 + koff + k] * Wk[k * 64 + d];
  ws[AMAT + (size_t)which * 4096 + rem] = s;
}

// scatter: segment sums + counts (native fp32 atomics; tables are L2-resident)
__global__ void k_scatter(const float* __restrict__ value,
                          const int* __restrict__ idx,
                          float* __restrict__ ws) {
  const size_t total = (size_t)Bn * En * Dn;
  size_t t = (size_t)blockIdx.x * blockDim.x + threadIdx.x;
  const size_t step = (size_t)gridDim.x * blockDim.x; // multiple of 64
  float* seg_row = ws + SEG_ROW;
  float* seg_col = ws + SEG_COL;
  float* cnt_row = ws + CNT_ROW;
  float* cnt_col = ws + CNT_COL;
  for (; t < total; t += step) {
    const size_t g = t >> 6;        // edge id in [0, B*E)
    const int d = (int)(t & 63);
    const int b = (int)(g / En);
    const int i0 = idx[2 * g];
    const int i1 = idx[2 * g + 1];
    const float v = value[t];
    atomAddF(&seg_row[((size_t)b * NSEG + i0) * 64 + d], v);
    atomAddF(&seg_col[((size_t)b * NSEG + i1) * 64 + d], v);
    if (d == 0) {
      atomAddF(&cnt_row[(size_t)b * NSEG + i0], 1.0f);
      atomAddF(&cnt_col[(size_t)b * NSEG + i1], 1.0f);
    }
  }
}

// d_b[o] = Wo1 b_self + Wo2 b_row + Wo3 b_col + Wo4*(Wglob*mean_b + bglob) + b_out
// (global sum recovered as the sum of row-segment sums)
__global__ void k_glob_bias(const float* __restrict__ W_out,
                            const float* __restrict__ b_self,
                            const float* __restrict__ b_row,
                            const float* __restrict__ b_col,
                            const float* __restrict__ W_glob,
                            const float* __restrict__ b_glob,
                            const float* __restrict__ b_out,
                            float* __restrict__ ws) {
  __shared__ float gmean[Bn][64];
  const int t = threadIdx.x; // 256 threads: (b, d/o)
  const int b = t >> 6;
  const int d = t & 63;
  // phase 1: per-batch mean of value over E via segment-row sums
  const float* seg_row = ws + SEG_ROW;
  float s = 0.0f;
  for (int sg = 0; sg < NSEG; ++sg)
    s += seg_row[((size_t)b * NSEG + sg) * 64 + d];
  gmean[b][d] = s * (1.0f / (float)En);
  __syncthreads();
  // phase 2: fold biases; o = d
  const int o = d;
  float acc = b_out[o];
#pragma unroll 4
  for (int k = 0; k < 64; ++k) {
    acc += W_out[o * 256 + k] * b_self[k];
    acc += W_out[o * 256 + 64 + k] * b_row[k];
    acc += W_out[o * 256 + 128 + k] * b_col[k];
    float gl = b_glob[k];
    for (int dd = 0; dd < 64; ++dd) gl += W_glob[k * 64 + dd] * gmean[b][dd];
    acc += W_out[o * 256 + 192 + k] * gl;
  }
  ws[DVEC + (size_t)b * 64 + o] = acc;
}

// segment transform (WMMA): seg <- M * (seg / (cnt + 1e-9)), in place.
// blockIdx.y: 0 = row tables with Bm, 1 = col tables with Cm.
__global__ void __launch_bounds__(256)
k_seg_transform(float* __restrict__ ws) {
  __shared__ __align__(16) float sM[64][68];
  __shared__ __align__(16) float sX[128][68];
  const int kind = blockIdx.y;
  float* seg = ws + (kind ? SEG_COL : SEG_ROW);
  const float* cnt = ws + (kind ? CNT_COL : CNT_ROW);
  const float* M = ws + (kind ? CMAT : BMAT);

  // cooperative load of fold matrix (64x64 -> 1024 float4)
  const float4* m4 = (const float4*)M;
#pragma unroll
  for (int r = 0; r < 4; ++r) {
    int f = threadIdx.x + 256 * r;
    int row = f >> 4, c4 = f & 15;
    *(float4*)&sM[row][c4 * 4] = m4[f];
  }
  __syncthreads();

  const int wv = threadIdx.x >> 5;
  const int lane = threadIdx.x & 31;
  const int tile = blockIdx.x * 8 + wv; // B * (NSEG/16) = 2500 tiles per kind
  if (tile < Bn * (NSEG / 16)) {
    const int b = tile / (NSEG / 16);
    const int s0 = (tile % (NSEG / 16)) * 16;
    // stage 16 mean vectors into this wave's LDS region
    const int row = lane & 15;
    const int cb = (lane >> 4) * 32;
    const float cn = cnt[(size_t)b * NSEG + s0 + row];
    const float rcp = 1.0f / (cn + 1e-9f);
    const float* src = seg + ((size_t)b * NSEG + s0 + row) * 64;
#pragma unroll
    for (int q = 0; q < 8; ++q) {
      float4 x = *(const float4*)&src[cb + q * 4];
      x.x *= rcp; x.y *= rcp; x.z *= rcp; x.w *= rcp;
      *(float4*)&sX[16 * wv + row][cb + q * 4] = x;
    }
    v8f acc[4];
#pragma unroll
    for (int nt = 0; nt < 4; ++nt)
#pragma unroll
      for (int j = 0; j < 8; ++j) acc[nt][j] = 0.0f;
    tile_matmul_16x64((const float(*)[68])&sX[16 * wv][0], sM, lane, acc);
    // write back (in place): element (m, n)
#pragma unroll
    for (int nt = 0; nt < 4; ++nt) {
      const int n = nt * 16 + (lane & 15);
#pragma unroll
      for (int j = 0; j < 8; ++j) {
        const int m = j + 8 * (lane >> 4);
        seg[((size_t)b * NSEG + s0 + m) * 64 + n] = acc[nt][j];
      }
    }
  }
}

// main edge kernel: 1 block = 128 edges (8 waves x 16-edge WMMA tiles)
__global__ void __launch_bounds__(256)
k_edges(const float* __restrict__ value,
        const int* __restrict__ idx,
        const float* __restrict__ ws,
        float* __restrict__ outF) {
  __shared__ __align__(16) float sA[64][68];
  __shared__ __align__(16) float sX[128][68];
  const float* Brow = ws + SEG_ROW;
  const float* Ccol = ws + SEG_COL;
  const float* dvec = ws + DVEC;

  // cooperative loads: A (1024 float4) + 128x64 value tile (2048 float4)
  const float4* a4 = (const float4*)(ws + AMAT);
#pragma unroll
  for (int r = 0; r < 4; ++r) {
    int f = threadIdx.x + 256 * r;
    int row = f >> 4, c4 = f & 15;
    *(float4*)&sA[row][c4 * 4] = a4[f];
  }
  const long long base = (long long)blockIdx.x * 128; // global edge id
  const float4* v4 = (const float4*)(value + (size_t)base * 64);
#pragma unroll
  for (int r = 0; r < 8; ++r) {
    int f = threadIdx.x + 256 * r;
    int row = f >> 4, c4 = f & 15;
    *(float4*)&sX[row][c4 * 4] = v4[(size_t)row * 16 + c4];
  }
  __syncthreads();

  const int wv = threadIdx.x >> 5;
  const int lane = threadIdx.x & 31;
  const int b = (int)(base / En); // E % 128 == 0: whole block same batch

  // gather indices for this lane's 8 output rows
  int i0[8], i1[8];
#pragma unroll
  for (int j = 0; j < 8; ++j) {
    const long long e = base + 16 * wv + (j + 8 * (lane >> 4));
    i0[j] = idx[2 * e];
    i1[j] = idx[2 * e + 1];
  }
  // init accumulators with Bm*row_mean + Cm*col_mean + d_b (gathered, L2 hits)
  v8f acc[4];
#pragma unroll
  for (int nt = 0; nt < 4; ++nt) {
    const int n = nt * 16 + (lane & 15);
    const float db = dvec[b * 64 + n];
#pragma unroll
    for (int j = 0; j < 8; ++j) {
      acc[nt][j] = Brow[((size_t)b * NSEG + i0[j]) * 64 + n] +
                   Ccol[((size_t)b * NSEG + i1[j]) * 64 + n] + db;
    }
  }
  // acc += V(16x64) * A^T   -- 64 v_wmma_f32_16x16x4_f32 per wave
  tile_matmul_16x64((const float(*)[68])&sX[16 * wv][0], sA, lane, acc);

  // leaky-relu, repack through LDS (sX already consumed; per-wave DS in-order)
#pragma unroll
  for (int nt = 0; nt < 4; ++nt) {
    const int n = nt * 16 + (lane & 15);
#pragma unroll
    for (int j = 0; j < 8; ++j) {
      const int m = j + 8 * (lane >> 4);
      float x = acc[nt][j];
      sX[16 * wv + m][n] = (x > 0.0f) ? x : 0.01f * x;
    }
  }
  __syncthreads();

  // fully-coalesced b128 stores: 128 edges x 64 floats
  float4* o4 = (float4*)(outF + (size_t)base * 64);
#pragma unroll
  for (int r = 0; r < 8; ++r) {
    int f = threadIdx.x + 256 * r;
    int row = f >> 4, c4 = f & 15;
    o4[(size_t)row * 16 + c4] = *(const float4*)&sX[row][c4 * 4];
  }
}

__global__ void k_copy_index(const int4* __restrict__ src, int4* __restrict__ dst,
                             size_t n4) {
  size_t i = (size_t)blockIdx.x * blockDim.x + threadIdx.x;
  size_t step = (size_t)gridDim.x * blockDim.x;
  for (; i < n4; i += step) dst[i] = src[i];
}

// ---------------------------------------------------------------------------
extern "C" void kernel_launch(void* const* d_in, const int* in_sizes, int n_in,
                              void* d_out, int out_size, void* d_ws, size_t ws_size,
                              hipStream_t stream) {
  (void)in_sizes; (void)n_in; (void)out_size; (void)ws_size;
  const int*   index  = (const int*)d_in[0];
  const float* value  = (const float*)d_in[1];
  const float* W_self = (const float*)d_in[2];
  const float* b_self = (const float*)d_in[3];
  const float* W_row  = (const float*)d_in[4];
  const float* b_row  = (const float*)d_in[5];
  const float* W_col  = (const float*)d_in[6];
  const float* b_col  = (const float*)d_in[7];
  const float* W_glob = (const float*)d_in[8];
  const float* b_glob = (const float*)d_in[9];
  const float* W_out  = (const float*)d_in[10];
  const float* b_out  = (const float*)d_in[11];

  float* ws = (float*)d_ws;
  int*   outIdx = (int*)d_out;                          // (index, out) tuple:
  float* outF = (float*)d_out + (size_t)2 * Bn * En;    // ints first, then f32

  k_zero<<<2048, 256, 0, stream>>>(ws, WS_FLOATS);
  k_fold_mats<<<48, 256, 0, stream>>>(W_out, W_self, W_row, W_col, ws);
  k_scatter<<<4096, 256, 0, stream>>>(value, index, ws);
  k_glob_bias<<<1, 256, 0, stream>>>(W_out, b_self, b_row, b_col,
                                     W_glob, b_glob, b_out, ws);
  {
    dim3 grid((Bn * (NSEG / 16) + 7) / 8, 2);
    k_seg_transform<<<grid, 256, 0, stream>>>(ws);
  }
  {
    const int nblocks = (int)((Bn * En) / 128); // 8192
    k_edges<<<nblocks, 256, 0, stream>>>(value, index, ws, outF);
  }
  k_copy_index<<<512, 256, 0, stream>>>((const int4*)index, (int4*)outIdx,
                                        (size_t)Bn * En * 2 / 4);
}